// ResonantFFNWithAttenuation_75892072121054
// MI455X (gfx1250) — compile-verified
//
#include <hip/hip_runtime.h>
#include <math.h>

#define D_MODEL 512
#define NUM_N   512
#define LUT     4096
#define TOKENS  1024
#define TPT     8                     // tokens per tile in reduce kernel
#define P       (512 * 512)           // elements per 512x512 plane
#define TWO_PI  6.283185307179586f

typedef float v2f __attribute__((ext_vector_type(2)));
typedef float v8f __attribute__((ext_vector_type(8)));

// Workspace layout (floats):
//   ws + 0*P : params4 [d][n] float4 = (scale, bshift, attn_cos, attn_sin)  (4*P floats)
//   ws + 4*P : projTc  [n][d]   proj_cos_w[d][n]
//   ws + 5*P : projTs  [n][d]   proj_sin_w[d][n]
//   ws + 6*P : cos_sum [t][n]   (2*P floats)
//   ws + 8*P : sin_sum [t][n]   (2*P floats)
// Total: 10*P floats = 10 MB.

__global__ __launch_bounds__(256)
void prep_kernel(const float* __restrict__ W, const float* __restrict__ B,
                 const float* __restrict__ ac, const float* __restrict__ as_,
                 const float* __restrict__ pc, const float* __restrict__ ps,
                 float* __restrict__ ws) {
    int idx = blockIdx.x * 256 + threadIdx.x;
    if (idx >= P) return;
    const float k = (float)LUT / TWO_PI;
    // W/B/attn are [n][d]
    int n = idx >> 9;           // idx / 512
    int d = idx & 511;
    float4* p4 = (float4*)ws;   // [d][n] packed
    float4 v;
    v.x = k / (1.0f + fabsf(W[idx]));
    v.y = B[idx] * k;
    v.z = ac[idx];
    v.w = as_[idx];
    p4[d * NUM_N + n] = v;
    // proj arrays are [d][n]; write projT[n][d]
    int dd = idx >> 9;
    int nn = idx & 511;
    int pdx = nn * D_MODEL + dd;
    ws[4 * P + pdx] = pc[idx];
    ws[5 * P + pdx] = ps[idx];
}

__global__ __launch_bounds__(256)
void reduce_kernel(const float* __restrict__ x, const float* __restrict__ ws,
                   const float* __restrict__ sin_t, const float* __restrict__ cos_t,
                   float* __restrict__ cos_sum, float* __restrict__ sin_sum) {
    __shared__ float lds_sin[LUT];
    __shared__ float lds_cos[LUT];
    __shared__ float lds_x[TPT * D_MODEL];

    int tid = threadIdx.x;
    for (int i = tid; i < LUT; i += 256) {
        lds_sin[i] = sin_t[i];
        lds_cos[i] = cos_t[i];
    }
    int t0 = blockIdx.x * TPT;
    for (int i = tid; i < TPT * D_MODEL; i += 256)
        lds_x[i] = x[t0 * D_MODEL + i];
    __syncthreads();

    int n = blockIdx.y * 256 + tid;
    const float4* p4 = (const float4*)ws;   // [d][n] packed params

    float cacc[TPT], sacc[TPT];
#pragma unroll
    for (int t = 0; t < TPT; ++t) { cacc[t] = 0.0f; sacc[t] = 0.0f; }

    for (int d = 0; d < D_MODEL; ++d) {
        float4 pr = p4[d * NUM_N + n];      // one coalesced b128 per d
        float sc  = pr.x;
        float bs  = pr.y;
        float wc  = pr.z;
        float wsn = pr.w;
#pragma unroll
        for (int t = 0; t < TPT; ++t) {
            float th = fmaf(lds_x[t * D_MODEL + d], sc, bs);   // same addr across wave -> broadcast
            int idx = __float2int_rn(th) & (LUT - 1);          // round-half-even + python mod
            cacc[t] = fmaf(lds_cos[idx], wc,  cacc[t]);
            sacc[t] = fmaf(lds_sin[idx], wsn, sacc[t]);
        }
    }
#pragma unroll
    for (int t = 0; t < TPT; ++t) {
        cos_sum[(t0 + t) * NUM_N + n] = cacc[t];
        sin_sum[(t0 + t) * NUM_N + n] = sacc[t];
    }
}

// One wave per 16x16 output tile. C[t][d] = sum_n cs[t][n]*projTc[n][d] + ss[t][n]*projTs[n][d]
__global__ __launch_bounds__(256)
void gemm_silu_kernel(const float* __restrict__ cos_sum, const float* __restrict__ sin_sum,
                      const float* __restrict__ projTc, const float* __restrict__ projTs,
                      float* __restrict__ out) {
    int wave = threadIdx.x >> 5;
    int lane = threadIdx.x & 31;
    int tile = blockIdx.x * 8 + wave;       // 2048 tiles: 64 token-tiles x 32 d-tiles
    int trow = (tile >> 5) * 16;            // token base
    int dcol = (tile & 31) * 16;            // d_model base
    int half = lane >> 4;                   // lane half selects K pair
    int l15  = lane & 15;

    const float* arow_c = cos_sum + (trow + l15) * NUM_N + 2 * half;
    const float* arow_s = sin_sum + (trow + l15) * NUM_N + 2 * half;

    v8f c = {};
    for (int k = 0; k < NUM_N; k += 4) {
        int kb = k + 2 * half;
        v2f a_c = { arow_c[k], arow_c[k + 1] };
        v2f b_c = { projTc[(kb + 0) * D_MODEL + dcol + l15],
                    projTc[(kb + 1) * D_MODEL + dcol + l15] };
        c = __builtin_amdgcn_wmma_f32_16x16x4_f32(false, a_c, false, b_c,
                                                  (short)0, c, false, false);
        v2f a_s = { arow_s[k], arow_s[k + 1] };
        v2f b_s = { projTs[(kb + 0) * D_MODEL + dcol + l15],
                    projTs[(kb + 1) * D_MODEL + dcol + l15] };
        c = __builtin_amdgcn_wmma_f32_16x16x4_f32(false, a_s, false, b_s,
                                                  (short)0, c, false, false);
    }

#pragma unroll
    for (int i = 0; i < 8; ++i) {
        int row = trow + i + 8 * half;      // C VGPR i, lane-half -> M
        float v = c[i];
        float sv = v / (1.0f + expf(-v));   // silu
        out[row * D_MODEL + dcol + l15] = sv;
    }
}

extern "C" void kernel_launch(void* const* d_in, const int* in_sizes, int n_in,
                              void* d_out, int out_size, void* d_ws, size_t ws_size,
                              hipStream_t stream) {
    const float* x   = (const float*)d_in[0];
    const float* W   = (const float*)d_in[1];
    const float* B   = (const float*)d_in[2];
    const float* ac  = (const float*)d_in[3];
    const float* as_ = (const float*)d_in[4];
    const float* pc  = (const float*)d_in[5];
    const float* ps  = (const float*)d_in[6];
    const float* st  = (const float*)d_in[7];
    const float* ct  = (const float*)d_in[8];
    float* ws  = (float*)d_ws;
    float* out = (float*)d_out;

    float* cos_sum = ws + 6 * (size_t)P;
    float* sin_sum = ws + 8 * (size_t)P;

    prep_kernel<<<(P + 255) / 256, 256, 0, stream>>>(W, B, ac, as_, pc, ps, ws);
    reduce_kernel<<<dim3(TOKENS / TPT, 2), 256, 0, stream>>>(x, ws, st, ct, cos_sum, sin_sum);
    gemm_silu_kernel<<<256, 256, 0, stream>>>(cos_sum, sin_sum,
                                              ws + 4 * (size_t)P, ws + 5 * (size_t)P, out);
}